// InterventionMamba_9663676416242
// MI455X (gfx1250) — compile-verified
//
#include <hip/hip_runtime.h>

// ---------------------------------------------------------------------------
// InterventionMamba fused kernel for MI455X (gfx1250, wave32, WMMA).
//
// Bu = (Bmat + dB) @ u is a bilinear form in u:
//   Bu[h] = sum_t o[t] * W2[h][t],  o = [vec(u u^T); u]  (t < 1056)
// where W2[h][t] = Ws[h*1024 + t] for t < 1024 (Ws natural layout!) and
// W2[h][1024+i] = Bmat[h*32+i] + bs[h*32+i].  So the heavy work is a K=1056
// f32 GEMM -> V_WMMA_F32_16X16X4_F32, with W2 (271 KB) resident in the
// 320 KB WGP LDS, staged by GLOBAL_LOAD_ASYNC_TO_LDS_B128 (coalesced copy).
// One workgroup (8 wave32) per batch row; h-recurrence lives in registers of
// threads 0..63.  `times` is produced by a separate coalesced transpose.
// ---------------------------------------------------------------------------

typedef float v2f __attribute__((ext_vector_type(2)));
typedef float v8f __attribute__((ext_vector_type(8)));

constexpr int kB  = 512;
constexpr int kK  = 1024;
constexpr int kDU = 32;
constexpr int kDH = 64;
constexpr int kNI = 5;
constexpr int kM  = 32;                    // events per chunk (2 WMMA M-tiles)
constexpr int kKdim = kDU * kDU + kDU;     // 1056 = quadratic + linear terms
constexpr int kW2Row = 1060;               // padded LDS row stride (16B align)
constexpr int kThreads = 256;              // 8 wave32

// ---- dynamic-LDS layout (float offsets) ----
constexpr int OFF_W2  = 0;                         // 64*1060 = 67840
constexpr int OFF_U   = OFF_W2 + kDH * kW2Row;     // 32*33 (padded rows)
constexpr int OFF_AB  = OFF_U + kM * 33;           // 32*64
constexpr int OFF_X   = OFF_AB + kM * kDH;         // 32*64
constexpr int OFF_WP  = OFF_X + kM * kDH;          // 32*5
constexpr int OFF_BP  = OFF_WP + kDU * kNI;        // 32
constexpr int OFF_G   = OFF_BP + kDU;              // 32
constexpr int OFF_BE  = OFF_G + kDU;               // 32
constexpr int OFF_A   = OFF_BE + kDU;              // 64   (A = -exp(A_log))
constexpr int OFF_WSL = OFF_A + kDH;               // 32   (Ws last row: delta)
constexpr int OFF_UIN = OFF_WSL + kDU;             // 32*5 staged raw inputs
constexpr int OFF_MU  = OFF_UIN + kM * kNI;        // 32
constexpr int OFF_RS  = OFF_MU + kM;               // 32
constexpr int OFF_DTD = OFF_RS + kM;               // 32   (dt * delta)
constexpr int OFF_ACT = OFF_DTD + kM;              // 32   (active mask)
constexpr int OFF_BSL = OFF_ACT + kM;              // 1    (bs last)
constexpr int kSmemFloats = OFF_BSL + 4;           // ~294.5 KB total

__global__ __launch_bounds__(kThreads)
void mamba_fused_kernel(const float* __restrict__ h0,
                        const float* __restrict__ t_int,
                        const float* __restrict__ U_int,
                        const int*   __restrict__ int_lens,
                        const float* __restrict__ Wp,
                        const float* __restrict__ bp,
                        const float* __restrict__ gamma,
                        const float* __restrict__ beta,
                        const float* __restrict__ A_log,
                        const float* __restrict__ Bmat,
                        const float* __restrict__ Ws,
                        const float* __restrict__ bs,
                        float* __restrict__ out_hfinal,
                        float* __restrict__ out_states)
{
    extern __shared__ float smem[];
    const int tid = threadIdx.x;
    const int b   = blockIdx.x;

    float* w2   = smem + OFF_W2;     // [64][1060], W2[h][t]
    float* us   = smem + OFF_U;      // layernormed u tile, rows padded to 33
    float* abar = smem + OFF_AB;
    float* xs   = smem + OFF_X;
    float* wp   = smem + OFF_WP;
    float* bps  = smem + OFF_BP;
    float* gs   = smem + OFF_G;
    float* bes  = smem + OFF_BE;
    float* As   = smem + OFF_A;
    float* wsl  = smem + OFF_WSL;
    float* uin  = smem + OFF_UIN;
    float* mus  = smem + OFF_MU;
    float* rss  = smem + OFF_RS;
    float* dtd  = smem + OFF_DTD;
    float* act  = smem + OFF_ACT;

    // ---- quadratic part of W2: flat coalesced Ws copy via async-to-LDS ----
    // W2[h][t] = Ws[h*1024 + t] for t<1024; LDS row stride 1060 floats.
    {
        const unsigned w2Base = (unsigned)(uintptr_t)(const void*)w2; // LDS off
        #pragma unroll 4
        for (int it = 0; it < (kDU * kDU * kDH) / (4 * kThreads); ++it) {
            const int e = (tid + it * kThreads) * 4;        // element idx, %4==0
            const int h = e >> 10, t = e & 1023;
            const unsigned ldsOff = w2Base + (unsigned)((h * kW2Row + t) * 4);
            const float* src = Ws + e;                       // 16B aligned
            asm volatile("global_load_async_to_lds_b128 %0, %1, off"
                         :: "v"(ldsOff), "v"(src) : "memory");
        }
    }

    // ---- linear part of W2 + small params (regular path, needs VALU) ----
    for (int idx = tid; idx < kDH * kDU; idx += kThreads) {
        const int h = idx >> 5, i = idx & 31;
        w2[h * kW2Row + kDU * kDU + i] = Bmat[idx] + bs[idx];
    }
    for (int idx = tid; idx < kDU * kNI; idx += kThreads) wp[idx] = Wp[idx];
    if (tid < kDU) {
        bps[tid] = bp[tid];
        gs[tid]  = gamma[tid];
        bes[tid] = beta[tid];
        wsl[tid] = Ws[(size_t)(kDH * kDU + kDH) * kDU + tid];   // row 2112
    }
    if (tid < kDH) As[tid] = -__expf(A_log[tid]);
    if (tid == 0)  smem[OFF_BSL] = bs[kDH * kDU + kDH];

    // ---- per-thread recurrence state (threads 0..63 own h-lanes) ----
    float hreg = 0.0f;
    if (tid < kDH) {
        hreg = h0[(size_t)b * kDH + tid];
        out_states[(size_t)b * kDH + tid] = hreg;               // states row 0
    }
    const int len = int_lens[b];

    // ---- wave / lane decomposition for WMMA ----
    const int lane = tid & 31;
    const int wave = tid >> 5;
    const int half = lane >> 4;          // K-half of the fragment
    const int nn   = lane & 15;
    const int m0   = (wave & 1) * 16;    // M-tile (event rows)
    const int hb   = (wave >> 1) * 16;   // N-tile (h columns)
    const float* wn = w2 + (hb + nn) * kW2Row;   // this lane's W2 row

    asm volatile("s_wait_asynccnt 0" ::: "memory");   // W2 quadratic part done
    __syncthreads();

    for (int k0 = 0; k0 < kK; k0 += kM) {
        __syncthreads();   // previous chunk fully consumed

        // ---- stage raw intervention inputs for this chunk ----
        for (int idx = tid; idx < kM * kNI; idx += kThreads)
            uin[idx] = U_int[((size_t)b * kK + k0 + idx / kNI) * kNI + idx % kNI];
        __syncthreads();

        // ---- projection 5 -> 32 ----
        for (int idx = tid; idx < kM * kDU; idx += kThreads) {
            const int r = idx >> 5, c = idx & 31;
            float p = bps[c];
            #pragma unroll
            for (int i = 0; i < kNI; ++i) p += uin[r * kNI + i] * wp[c * kNI + i];
            us[r * 33 + c] = p;
        }
        __syncthreads();

        // ---- layernorm stats ----
        if (tid < kM) {
            float m = 0.0f;
            #pragma unroll
            for (int c = 0; c < kDU; ++c) m += us[tid * 33 + c];
            m *= (1.0f / kDU);
            float v = 0.0f;
            #pragma unroll
            for (int c = 0; c < kDU; ++c) {
                const float d = us[tid * 33 + c] - m;
                v += d * d;
            }
            v *= (1.0f / kDU);
            mus[tid] = m;
            rss[tid] = rsqrtf(v + 1e-5f);
        }
        __syncthreads();

        // ---- apply layernorm in place ----
        for (int idx = tid; idx < kM * kDU; idx += kThreads) {
            const int r = idx >> 5, c = idx & 31;
            us[r * 33 + c] = (us[r * 33 + c] - mus[r]) * rss[r] * gs[c] + bes[c];
        }
        __syncthreads();

        // ---- dt, delta = softplus(u . ws_last + bs_last), active mask ----
        if (tid < kM) {
            const int k = k0 + tid;
            const float tk = t_int[(size_t)b * kK + k];
            const float d  = (k == 0) ? 0.0f
                                      : fmaxf(tk - t_int[(size_t)b * kK + k - 1], 0.0f);
            float s = smem[OFF_BSL];
            #pragma unroll
            for (int i = 0; i < kDU; ++i) s += us[tid * 33 + i] * wsl[i];
            const float delta = (s > 20.0f) ? s : log1pf(__expf(s));
            dtd[tid] = d * delta;
            act[tid] = (k < len) ? 1.0f : 0.0f;
        }

        // ---- GEMM: Bu(32x64) = O(32x1056) @ W2^T, f32 WMMA, dual acc ----
        const float* ur = us + (m0 + nn) * 33;   // this lane's event row
        v8f acc0 = {0.f, 0.f, 0.f, 0.f, 0.f, 0.f, 0.f, 0.f};
        v8f acc1 = {0.f, 0.f, 0.f, 0.f, 0.f, 0.f, 0.f, 0.f};

        #pragma unroll 4
        for (int kk = 0; kk < kDU * kDU; kk += 8) {     // quadratic terms
            const int t0 = kk + half * 2;
            const int i  = t0 >> 5, j = t0 & 31;
            const float ui = ur[i];
            v2f a0; a0.x = ui * ur[j];     a0.y = ui * ur[j + 1];
            v2f a1; a1.x = ui * ur[j + 4]; a1.y = ui * ur[j + 5];
            const v2f b0 = *(const v2f*)(wn + t0);       // ds_load_b64
            const v2f b1 = *(const v2f*)(wn + t0 + 4);
            acc0 = __builtin_amdgcn_wmma_f32_16x16x4_f32(
                false, a0, false, b0, (short)0, acc0, false, false);
            acc1 = __builtin_amdgcn_wmma_f32_16x16x4_f32(
                false, a1, false, b1, (short)0, acc1, false, false);
        }
        #pragma unroll
        for (int kk = kDU * kDU; kk < kKdim; kk += 8) { // linear terms
            const int t0 = kk + half * 2;
            const int i0 = t0 - kDU * kDU;
            v2f a0; a0.x = ur[i0];     a0.y = ur[i0 + 1];
            v2f a1; a1.x = ur[i0 + 4]; a1.y = ur[i0 + 5];
            const v2f b0 = *(const v2f*)(wn + t0);
            const v2f b1 = *(const v2f*)(wn + t0 + 4);
            acc0 = __builtin_amdgcn_wmma_f32_16x16x4_f32(
                false, a0, false, b0, (short)0, acc0, false, false);
            acc1 = __builtin_amdgcn_wmma_f32_16x16x4_f32(
                false, a1, false, b1, (short)0, acc1, false, false);
        }
        const v8f accs = acc0 + acc1;
        __syncthreads();   // dtd/act visible; GEMM done

        // ---- A_bar, X = (A_bar-1)/A * Bu  (per D-fragment element) ----
        #pragma unroll
        for (int v = 0; v < 8; ++v) {
            const int rr = m0 + v + half * 8;
            const int hh = hb + nn;
            const float Ah = As[hh];
            const float ab = __expf(Ah * dtd[rr]);
            abar[rr * kDH + hh] = ab;
            xs[rr * kDH + hh]   = (ab - 1.0f) / Ah * accs[v];
        }
        __syncthreads();

        // ---- sequential recurrence over the 32 events of this chunk ----
        if (tid < kDH) {
            for (int rr = 0; rr < kM; ++rr) {
                if (act[rr] != 0.0f)
                    hreg = abar[rr * kDH + tid] * hreg + xs[rr * kDH + tid];
                out_states[(size_t)(k0 + rr + 1) * kB * kDH + (size_t)b * kDH + tid] = hreg;
            }
        }
    }

    if (tid < kDH) out_hfinal[(size_t)b * kDH + tid] = hreg;
}

// ---- coalesced transpose producing times = [t_int[:,0]; t_int.T] ----
__global__ __launch_bounds__(256)
void times_transpose_kernel(const float* __restrict__ t_int,
                            float* __restrict__ out_times)
{
    __shared__ float tile[32][33];
    const int tx = threadIdx.x & 31;
    const int ty = threadIdx.x >> 5;
    const int k0 = blockIdx.x * 32;
    const int b0 = blockIdx.y * 32;

    #pragma unroll
    for (int i = ty; i < 32; i += 8)
        tile[i][tx] = t_int[(size_t)(b0 + i) * kK + k0 + tx];   // coalesced read
    __syncthreads();
    #pragma unroll
    for (int i = ty; i < 32; i += 8)
        out_times[(size_t)(k0 + i + 1) * kB + b0 + tx] = tile[tx][i]; // coalesced
    if (k0 == 0 && ty == 0)                                     // row 0
        out_times[b0 + tx] = tile[tx][0];
}

extern "C" void kernel_launch(void* const* d_in, const int* in_sizes, int n_in,
                              void* d_out, int out_size, void* d_ws, size_t ws_size,
                              hipStream_t stream) {
    (void)in_sizes; (void)n_in; (void)out_size; (void)d_ws; (void)ws_size;

    const float* h0       = (const float*)d_in[0];
    const float* t_int    = (const float*)d_in[1];
    const float* U_int    = (const float*)d_in[2];
    const int*   int_lens = (const int*)  d_in[3];
    const float* Wp       = (const float*)d_in[4];
    const float* bp       = (const float*)d_in[5];
    const float* gamma    = (const float*)d_in[6];
    const float* beta     = (const float*)d_in[7];
    const float* A_log    = (const float*)d_in[8];
    const float* Bmat     = (const float*)d_in[9];
    const float* Ws       = (const float*)d_in[10];
    const float* bs       = (const float*)d_in[11];

    float* out        = (float*)d_out;
    float* out_hfinal = out;                                  // (B, 64)
    float* out_times  = out + (size_t)kB * kDH;               // (K+1, B)
    float* out_states = out_times + (size_t)(kK + 1) * kB;    // (K+1, B, 64)

    const size_t smem = (size_t)kSmemFloats * sizeof(float);  // ~294.5 KB
    (void)hipFuncSetAttribute((const void*)mamba_fused_kernel,
                              hipFuncAttributeMaxDynamicSharedMemorySize,
                              (int)smem);

    times_transpose_kernel<<<dim3(kK / 32, kB / 32), 256, 0, stream>>>(
        t_int, out_times);

    mamba_fused_kernel<<<kB, kThreads, smem, stream>>>(
        h0, t_int, U_int, int_lens, Wp, bp, gamma, beta, A_log, Bmat, Ws, bs,
        out_hfinal, out_states);
}